// MixedOp_34832184770683
// MI455X (gfx1250) — compile-verified
//
#include <hip/hip_runtime.h>

// MI455X / gfx1250, wave32. One workgroup (128 thr = 4 waves) per (b,c) SPD problem.
// All small-matrix algebra done with V_WMMA_F32_16X16X4_F32 (fp32 matrix cores,
// highest-precision WMMA on CDNA5), operands resident in LDS.

typedef __attribute__((ext_vector_type(2))) float v2f;
typedef __attribute__((ext_vector_type(8))) float v8f;

// C[M x N] = A[M x K] * B[K x N]   (B optionally transposed: B is [N x K] row-major)
// M,N multiples of 16; tiles distributed over the 4 waves; K multiple of 4.
// Leading barriers make producer LDS writes visible; trailing barrier publishes C.
__device__ __forceinline__ void mm(float* __restrict__ C, int ldc,
                                   const float* __restrict__ A, int lda,
                                   const float* __restrict__ B, int ldb,
                                   int M, int N, int Kd, bool transB)
{
    __syncthreads();
    const int wave = threadIdx.x >> 5;
    const int lane = threadIdx.x & 31;
    const int half = lane >> 4;       // lanes 16-31 handle K+2/K+3 (A) and M+8 rows (C)
    const int l15  = lane & 15;
    const int tn = N >> 4;
    const int tiles = (M >> 4) * tn;  // always multiple of 4 here -> uniform trip count
    for (int t = wave; t < tiles; t += 4) {
        const int mi = (t / tn) << 4;
        const int ni = (t % tn) << 4;
        v8f acc = {};
        #pragma unroll 4
        for (int k = 0; k < Kd; k += 4) {
            v2f a, b;
            const float* ar = A + (mi + l15) * lda + k + 2 * half;   // 16x4 A frag
            a.x = ar[0];
            a.y = ar[1];
            if (!transB) {                                           // 4x16 B frag
                const float* br = B + (k + 2 * half) * ldb + ni + l15;
                b.x = br[0];
                b.y = br[ldb];
            } else {
                const float* br = B + (ni + l15) * ldb + k + 2 * half;
                b.x = br[0];
                b.y = br[1];
            }
            acc = __builtin_amdgcn_wmma_f32_16x16x4_f32(
                false, a, false, b, (short)0, acc, false, false);
        }
        #pragma unroll
        for (int r = 0; r < 8; ++r)
            C[(mi + half * 8 + r) * ldc + ni + l15] = acc[r];
    }
    __syncthreads();
}

// trace of a 32x32 matrix in LDS, broadcast to all threads
__device__ __forceinline__ float trace32(const float* A, float* red)
{
    __syncthreads();
    if (threadIdx.x < 32) red[threadIdx.x] = A[threadIdx.x * 33];
    __syncthreads();
    float s = 0.f;
    #pragma unroll
    for (int i = 0; i < 32; ++i) s += red[i];
    return s;
}

// Coupled Newton-Schulz: src SPD (32x32) -> dst_s = src^{1/2}, dst_i = src^{-1/2}.
// Trace pre-scaling guarantees spectrum in (0,1] -> NS converges.
// y,z,t,yn,zn: five distinct 1024-float LDS scratch buffers.
__device__ void ns_sqrt(float* dst_s, float* dst_i, const float* src,
                        float* y, float* z, float* t, float* yn, float* zn,
                        float* red, int iters)
{
    const int tid = threadIdx.x;
    const float c = trace32(src, red);
    const float inv_c = 1.f / c;
    __syncthreads();
    for (int i = tid; i < 1024; i += 128) {
        y[i] = src[i] * inv_c;
        z[i] = ((i % 33) == 0) ? 1.f : 0.f;
    }
    for (int it = 0; it < iters; ++it) {
        mm(t, 32, z, 32, y, 32, 32, 32, 32, false);         // t = Z*Y
        for (int i = tid; i < 1024; i += 128)               // t = (3I - ZY)/2
            t[i] = (((i % 33) == 0) ? 1.5f : 0.f) - 0.5f * t[i];
        mm(yn, 32, y, 32, t, 32, 32, 32, 32, false);        // Y <- Y*t
        mm(zn, 32, t, 32, z, 32, 32, 32, 32, false);        // Z <- t*Z
        float* tmp;
        tmp = y; y = yn; yn = tmp;
        tmp = z; z = zn; zn = tmp;
    }
    const float sc  = __fsqrt_rn(c);
    const float isc = __frsqrt_rn(c);
    __syncthreads();
    if (dst_s) for (int i = tid; i < 1024; i += 128) dst_s[i] = y[i] * sc;
    if (dst_i) for (int i = tid; i < 1024; i += 128) dst_i[i] = z[i] * isc;
    __syncthreads();
}

__global__ __launch_bounds__(128, 1)
void spdnet_mixedop_kernel(const float* __restrict__ x,    // [2048,64,64]
                           const float* __restrict__ wts,  // [4]
                           const float* __restrict__ Ws,   // [4,32,64]
                           float* __restrict__ out)        // [2048,32,32]
{
    __shared__ float sX[4096];            // x (64x64); later: Sq | Si | U | free
    __shared__ float sWk[2048];           // W_k (32x64); later generic scratch x2
    __shared__ float sP[2048];            // W_k*x (32x64); later generic scratch x2
    __shared__ float sS[4][1024];         // s_k = W_k x W_k^T
    __shared__ float sB0[1024], sB1[1024], sB2[1024];
    __shared__ float sRed[32];

    const int bc  = blockIdx.x;
    const int tid = threadIdx.x;

    float wk[4];
    #pragma unroll
    for (int k = 0; k < 4; ++k) wk[k] = wts[k];

    for (int i = tid; i < 4096; i += 128) sX[i] = x[bc * 4096 + i];

    // ---- Phase A: s_k = W_k x W_k^T (32x64x64 + 32x64x32 WMMA GEMMs) ----
    for (int k = 0; k < 4; ++k) {
        __syncthreads();
        for (int i = tid; i < 2048; i += 128) sWk[i] = Ws[k * 2048 + i];
        mm(sP, 64, sWk, 64, sX, 64, 32, 64, 64, false);    // P = W_k * x
        mm(sS[k], 32, sP, 64, sWk, 64, 32, 32, 64, true);  // s_k = P * W_k^T
    }

    // ---- Phase B: G0 = sum w_k s_k ; Sq = G0^{1/2}, Si = G0^{-1/2} ----
    __syncthreads();
    for (int i = tid; i < 1024; i += 128)
        sB0[i] = wk[0] * sS[0][i] + wk[1] * sS[1][i] +
                 wk[2] * sS[2][i] + wk[3] * sS[3][i];

    float* pSq = sX;              // x is dead: repurpose sX
    float* pSi = sX + 1024;
    float* pU  = sX + 2048;
    ns_sqrt(pSq, pSi, sB0, sB1, sB2, sWk, sWk + 1024, sP, sRed, 14);

    __syncthreads();
    for (int i = tid; i < 1024; i += 128) pU[i] = 0.f;

    // ---- Phase C: per k, T_k = Si s_k Si ; L_k = logm(T_k) ; U += w_k L_k ----
    for (int k = 0; k < 4; ++k) {
        mm(sB0, 32, pSi, 32, sS[k], 32, 32, 32, 32, false);   // Si * s_k
        mm(sB1, 32, sB0, 32, pSi, 32, 32, 32, 32, false);     // T_k

        // inverse scaling: 3 matrix square roots -> M = T_k^(1/8) near I
        ns_sqrt(sB2, nullptr, sB1, sP, sP + 1024, sWk, sWk + 1024, sB0, sRed, 12);
        ns_sqrt(sB1, nullptr, sB2, sP, sP + 1024, sWk, sWk + 1024, sB0, sRed, 12);
        ns_sqrt(sB2, nullptr, sB1, sP, sP + 1024, sWk, sWk + 1024, sB0, sRed, 12);

        // E = M - I  (in sB2)
        __syncthreads();
        for (int i = tid; i < 1024; i += 128)
            if ((i % 33) == 0) sB2[i] -= 1.f;

        // Gregory/Horner: log(I+E) = E*(a1 + E*(a2 + ... )), a_i = (-1)^{i+1}/i
        float* q  = sB0;
        float* qn = sB1;
        __syncthreads();
        for (int i = tid; i < 1024; i += 128)
            q[i] = ((i % 33) == 0) ? (-1.f / 8.f) : 0.f;      // a8*I
        for (int ii = 7; ii >= 1; --ii) {
            mm(qn, 32, sB2, 32, q, 32, 32, 32, 32, false);    // E*q
            const float a = ((ii & 1) ? 1.f : -1.f) / (float)ii;
            for (int i = tid; i < 1024; i += 128)
                if ((i % 33) == 0) qn[i] += a;
            float* tmp = q; q = qn; qn = tmp;
        }
        mm(sWk, 32, sB2, 32, q, 32, 32, 32, 32, false);       // L = E*q

        const float scale = wk[k] * 8.f;                      // * 2^3 (undo scaling)
        for (int i = tid; i < 1024; i += 128) pU[i] += scale * sWk[i];
        __syncthreads();
    }

    // ---- Phase D: out = Sq * expm(U) * Sq  (scaling & squaring, m=6) ----
    __syncthreads();
    for (int i = tid; i < 1024; i += 128) {
        sB0[i] = pU[i] * (1.f / 64.f);                        // V = U/2^6
        sB1[i] = ((i % 33) == 0) ? 1.f : 0.f;                 // R = I
    }
    for (int ii = 10; ii >= 1; --ii) {                        // R = I + (V*R)/ii
        mm(sB2, 32, sB0, 32, sB1, 32, 32, 32, 32, false);
        const float inv = 1.f / (float)ii;
        for (int i = tid; i < 1024; i += 128)
            sB1[i] = (((i % 33) == 0) ? 1.f : 0.f) + sB2[i] * inv;
    }
    float* r  = sB1;
    float* rn = sB2;
    for (int sq = 0; sq < 6; ++sq) {                          // 6 squarings
        mm(rn, 32, r, 32, r, 32, 32, 32, 32, false);
        float* tmp = r; r = rn; rn = tmp;
    }
    mm(sB0, 32, pSq, 32, r, 32, 32, 32, 32, false);           // Sq * expm(U)
    mm(sP, 32, sB0, 32, pSq, 32, 32, 32, 32, false);          // ... * Sq
    __syncthreads();
    for (int i = tid; i < 1024; i += 128) out[bc * 1024 + i] = sP[i];
}

extern "C" void kernel_launch(void* const* d_in, const int* in_sizes, int n_in,
                              void* d_out, int out_size, void* d_ws, size_t ws_size,
                              hipStream_t stream)
{
    const float* x   = (const float*)d_in[0];   // [128,16,64,64]
    const float* wts = (const float*)d_in[1];   // [4]
    const float* Ws  = (const float*)d_in[2];   // [4,32,64]
    float* out = (float*)d_out;                 // [128,16,32,32]

    dim3 grid(2048);   // B*C independent SPD problems
    dim3 block(128);   // 4 waves of 32
    hipLaunchKernelGGL(spdnet_mixedop_kernel, grid, block, 0, stream,
                       x, wts, Ws, out);
}